// GCNLayer_5360119185954
// MI455X (gfx1250) — compile-verified
//
#include <hip/hip_runtime.h>
#include <hip/hip_bf16.h>

#define HID 16
#define HEADS 8

typedef __attribute__((ext_vector_type(2))) float v2f;
typedef __attribute__((ext_vector_type(8))) float v8f;

// ---------------- zero scratch ----------------
__global__ void zero_f4_kernel(float4* __restrict__ p, long n4) {
  long i = (long)blockIdx.x * blockDim.x + threadIdx.x;
  long stride = (long)gridDim.x * blockDim.x;
  float4 z = make_float4(0.f, 0.f, 0.f, 0.f);
  for (; i < n4; i += stride) p[i] = z;
}

// ---------------- GEMM: C[M,128] = A[M,128] @ B[128,128] + bias[128] ----------
// One block = 16 rows of A staged in LDS (padded stride 132 -> conflict-free
// ds_load_b64 for A fragments). 8 wave32 waves each own one 16x16 column tile
// and run 32 chained V_WMMA_F32_16X16X4_F32 ops over K=128.
__global__ __launch_bounds__(256) void gemm128_bias_kernel(
    const float* __restrict__ A, const float* __restrict__ B,
    const float* __restrict__ bias, float* __restrict__ C, int M)
{
  __shared__ float As[16 * 132];
  const int tid = threadIdx.x;
  const int m0 = blockIdx.x * 16;

  // cooperative load of the 16x128 A tile
  for (int i = tid; i < 16 * 128; i += 256) {
    int r = i >> 7, c = i & 127;
    int m = m0 + r;
    As[r * 132 + c] = (m < M) ? A[(long)m * 128 + c] : 0.f;
  }
  __syncthreads();

  const int lane = tid & 31;
  const int wave = tid >> 5;      // 0..7 -> column tile
  const int n0 = wave * 16;
  const int rc = lane & 15;       // row of A frag == col of B/C frag
  const int kof = (lane >> 4) << 1;  // lanes 16-31 hold K+2,K+3

  const float* Arow = As + rc * 132;
  const float* Bcol = B + n0 + rc;

  v8f acc = {};
#pragma unroll
  for (int k = 0; k < 128; k += 4) {
    v2f a, b;
    a.x = Arow[k + kof];
    a.y = Arow[k + kof + 1];
    b.x = Bcol[(k + kof) * 128];
    b.y = Bcol[(k + kof + 1) * 128];
    acc = __builtin_amdgcn_wmma_f32_16x16x4_f32(
        /*neg_a=*/false, a, /*neg_b=*/false, b,
        /*c_mod=*/(short)0, acc, /*reuse_a=*/false, /*reuse_b=*/false);
  }

  const float bv = bias[n0 + rc];
  const int mbase = m0 + ((lane >> 4) << 3);  // lanes 16-31 hold rows M=8..15
#pragma unroll
  for (int i = 0; i < 8; ++i) {
    int m = mbase + i;
    if (m < M) C[(long)m * 128 + n0 + rc] = acc[i] + bv;
  }
}

// ---------------- edge scatter: agg[dst] += hf[src] * norm[src] -------------
// One wave32 per edge: each lane handles one float4 (32 * 16B = 512B row).
__global__ __launch_bounds__(256) void edge_scatter_kernel(
    const float* __restrict__ hf, const int* __restrict__ src,
    const int* __restrict__ dst, const float* __restrict__ norm,
    float* __restrict__ agg, int nEdges)
{
  int e = blockIdx.x * 8 + (threadIdx.x >> 5);
  if (e >= nEdges) return;
  int lane = threadIdx.x & 31;
  int s = src[e];
  int d = dst[e];
  float ns = norm[s];
  float4 v = ((const float4*)(hf + (long)s * 128))[lane];
  float* out = agg + (long)d * 128 + lane * 4;
  __hip_atomic_fetch_add(out + 0, v.x * ns, __ATOMIC_RELAXED, __HIP_MEMORY_SCOPE_AGENT);
  __hip_atomic_fetch_add(out + 1, v.y * ns, __ATOMIC_RELAXED, __HIP_MEMORY_SCOPE_AGENT);
  __hip_atomic_fetch_add(out + 2, v.z * ns, __ATOMIC_RELAXED, __HIP_MEMORY_SCOPE_AGENT);
  __hip_atomic_fetch_add(out + 3, v.w * ns, __ATOMIC_RELAXED, __HIP_MEMORY_SCOPE_AGENT);
}

// ---------------- attention + gating --------------------------------------
// Faithful torch reshape [N,128]->(8,N,16): chunk g covers node g/8,
// channels 16*(g%8).., head g/N. Elementwise per 16-float chunk.
__global__ __launch_bounds__(256) void finalize_kernel(
    const float* __restrict__ hf, const float* __restrict__ agg1,
    const float* __restrict__ agg2, const float* __restrict__ norm1,
    const float* __restrict__ norm2, const float* __restrict__ al,
    const float* __restrict__ ar, float* __restrict__ comb, int N)
{
  int g = blockIdx.x * blockDim.x + threadIdx.x;
  if (g >= HEADS * N) return;
  int head = g / N;
  int node = g >> 3;
  int cbase = (g & 7) * HID;
  long off = (long)node * 128 + cbase;

  float n1 = norm1[node];
  float n2 = norm2[node];

  float hv[HID], h1v[HID], h2v[HID];
  float ai = 0.f, aj1 = 0.f, aj2 = 0.f;

  const float4* hp  = (const float4*)(hf + off);
  const float4* p1  = (const float4*)(agg1 + off);
  const float4* p2  = (const float4*)(agg2 + off);
  const float4* alp = (const float4*)(al + head * HID);
  const float4* arp = (const float4*)(ar + head * HID);

#pragma unroll
  for (int q = 0; q < 4; ++q) {
    float4 th = hp[q], t1 = p1[q], t2 = p2[q], ta = alp[q], tr = arp[q];
    float hq[4]  = {th.x, th.y, th.z, th.w};
    float a1q[4] = {t1.x, t1.y, t1.z, t1.w};
    float a2q[4] = {t2.x, t2.y, t2.z, t2.w};
    float alq[4] = {ta.x, ta.y, ta.z, ta.w};
    float arq[4] = {tr.x, tr.y, tr.z, tr.w};
#pragma unroll
    for (int j = 0; j < 4; ++j) {
      int d = q * 4 + j;
      hv[d] = hq[j];
      h1v[d] = a1q[j] * n1;
      h2v[d] = a2q[j] * n2;
      ai  += hv[d] * alq[j];
      aj1 += h1v[d] * arq[j];
      aj2 += h2v[d] * arq[j];
    }
  }

  float x1 = ai + aj1;
  x1 = x1 > 0.f ? x1 : 0.2f * x1;
  float x2 = ai + aj2;
  x2 = x2 > 0.f ? x2 : 0.2f * x2;
  // clip(exp(x), -10, 10): exp >= 0 so only the upper bound binds
  float e1 = fminf(__expf(x1), 10.f);
  float e2 = fminf(__expf(x2), 10.f);
  float inv = 1.f / (e1 + e2);
  float w1 = e1 * inv;
  float w2 = e2 * inv;

  float* op = comb + off;
#pragma unroll
  for (int d = 0; d < HID; ++d) op[d] = w1 * h1v[d] + w2 * h2v[d];
}

extern "C" void kernel_launch(void* const* d_in, const int* in_sizes, int n_in,
                              void* d_out, int out_size, void* d_ws, size_t ws_size,
                              hipStream_t stream) {
  const float* h     = (const float*)d_in[0];
  const int*   src1  = (const int*)d_in[1];
  const int*   dst1  = (const int*)d_in[2];
  const int*   src2  = (const int*)d_in[3];
  const int*   dst2  = (const int*)d_in[4];
  const float* norm1 = (const float*)d_in[5];
  const float* norm2 = (const float*)d_in[6];
  const float* W_lin = (const float*)d_in[7];
  const float* b_lin = (const float*)d_in[8];
  const float* al    = (const float*)d_in[9];
  const float* ar    = (const float*)d_in[10];
  const float* W_fc  = (const float*)d_in[11];
  const float* b_fc  = (const float*)d_in[12];
  float* out = (float*)d_out;

  const int N  = in_sizes[0] / 128;
  const int E1 = in_sizes[1];
  const int E2 = in_sizes[3];

  float* hf   = (float*)d_ws;
  float* agg1 = hf + (long)N * 128;
  float* agg2 = agg1 + (long)N * 128;
  float* comb = agg2 + (long)N * 128;

  // zero both aggregation buffers (contiguous)
  long n4 = (long)N * 128 * 2 / 4;
  zero_f4_kernel<<<2048, 256, 0, stream>>>((float4*)agg1, n4);

  // hf = h @ W_lin + b_lin
  gemm128_bias_kernel<<<(N + 15) / 16, 256, 0, stream>>>(h, W_lin, b_lin, hf, N);

  // two propagations (scatter-add side; dst-side norm folded into finalize)
  edge_scatter_kernel<<<(E1 + 7) / 8, 256, 0, stream>>>(hf, src1, dst1, norm1, agg1, E1);
  edge_scatter_kernel<<<(E2 + 7) / 8, 256, 0, stream>>>(hf, src2, dst2, norm2, agg2, E2);

  // attention + gating -> comb
  finalize_kernel<<<(HEADS * N + 255) / 256, 256, 0, stream>>>(
      hf, agg1, agg2, norm1, norm2, al, ar, comb, N);

  // out = comb @ W_fc + b_fc
  gemm128_bias_kernel<<<(N + 15) / 16, 256, 0, stream>>>(comb, W_fc, b_fc, out, N);
}